// GAT_50500225466774
// MI455X (gfx1250) — compile-verified
//
#include <hip/hip_runtime.h>

#define HH 8
#define CC 4
#define FF 32     // H*C
#define HIDN 64
#define NB 64     // batches

typedef __attribute__((ext_vector_type(2))) float v2f;
typedef __attribute__((ext_vector_type(8))) float v8f;

__device__ __forceinline__ unsigned ordEnc(float f) {
    unsigned b = __float_as_uint(f);
    return (b & 0x80000000u) ? ~b : (b | 0x80000000u);
}
__device__ __forceinline__ float ordDec(unsigned o) {
    unsigned b = (o & 0x80000000u) ? (o ^ 0x80000000u) : ~o;
    return __uint_as_float(b);
}
__device__ __forceinline__ float lrelu(float v) { return v > 0.f ? v : 0.2f * v; }

// ---- edge degree + loop edge-attr sums -------------------------------------
__global__ void k_deg(const int* __restrict__ dst, const float* __restrict__ ea,
                      float* __restrict__ deg, float* __restrict__ loopea, int E) {
    int e = blockIdx.x * blockDim.x + threadIdx.x;
    if (e >= E) return;
    int d = dst[e];
    atomicAdd(&deg[d], 1.0f);
    atomicAdd(&loopea[d], ea[e]);
}

// ---- xl / xr projection + finalize loop_ea ---------------------------------
__global__ void k_node_prep(const float* __restrict__ x,
                            const float* __restrict__ wl, const float* __restrict__ bl,
                            const float* __restrict__ wr, const float* __restrict__ br,
                            const float* __restrict__ deg,
                            float* __restrict__ xl, float* __restrict__ xr,
                            float* __restrict__ loopea, int N) {
    int t = blockIdx.x * blockDim.x + threadIdx.x;
    if (t >= N * FF) return;
    int i = t / FF, f = t % FF;
    float xv = x[i];
    xl[t] = xv * wl[f] + bl[f];
    xr[t] = xv * wr[f] + br[f];
    if (f == 0) loopea[i] = loopea[i] / fmaxf(deg[i], 1.0f);
}

// ---- pass A: segment max of alpha over augmented edges ---------------------
__global__ void k_alpha_max(const int* __restrict__ src, const int* __restrict__ dst,
                            const float* __restrict__ eattr, const float* __restrict__ loopea,
                            const float* __restrict__ xl, const float* __restrict__ xr,
                            const float* __restrict__ we, const float* __restrict__ att,
                            unsigned* __restrict__ mord, int E, int N) {
    int t = blockIdx.x * blockDim.x + threadIdx.x;
    if (t >= (E + N) * HH) return;
    int e = t / HH, h = t % HH;
    int s, d; float ea;
    if (e < E) { s = src[e]; d = dst[e]; ea = eattr[e]; }
    else       { s = d = e - E; ea = loopea[s]; }
    const float* xls = xl + (size_t)s * FF + h * CC;
    const float* xrd = xr + (size_t)d * FF + h * CC;
    float alpha = 0.f;
#pragma unroll
    for (int c = 0; c < CC; ++c)
        alpha += att[h * CC + c] * lrelu(xls[c] + xrd[c] + ea * we[h * CC + c]);
    atomicMax(&mord[(size_t)d * HH + h], ordEnc(alpha));
}

// ---- pass B: exp / denom / weighted message accumulation -------------------
__global__ void k_soft_acc(const int* __restrict__ src, const int* __restrict__ dst,
                           const float* __restrict__ eattr, const float* __restrict__ loopea,
                           const float* __restrict__ xl, const float* __restrict__ xr,
                           const float* __restrict__ we, const float* __restrict__ att,
                           const unsigned* __restrict__ mord,
                           float* __restrict__ den, float* __restrict__ outacc,
                           int E, int N) {
    int t = blockIdx.x * blockDim.x + threadIdx.x;
    if (t >= (E + N) * HH) return;
    int e = t / HH, h = t % HH;
    int s, d; float ea;
    if (e < E) { s = src[e]; d = dst[e]; ea = eattr[e]; }
    else       { s = d = e - E; ea = loopea[s]; }
    const float* xls = xl + (size_t)s * FF + h * CC;
    const float* xrd = xr + (size_t)d * FF + h * CC;
    float alpha = 0.f;
#pragma unroll
    for (int c = 0; c < CC; ++c)
        alpha += att[h * CC + c] * lrelu(xls[c] + xrd[c] + ea * we[h * CC + c]);
    float m = ordDec(mord[(size_t)d * HH + h]);
    float ex = expf(alpha - m);
    atomicAdd(&den[(size_t)d * HH + h], ex);
    float* op = outacc + (size_t)d * FF + h * CC;
#pragma unroll
    for (int c = 0; c < CC; ++c) atomicAdd(&op[c], xls[c] * ex);
}

// ---- h = relu(out/den + bias); accumulate per-graph sums -------------------
__global__ void k_h_relu(const float* __restrict__ outacc, const float* __restrict__ den,
                         const float* __restrict__ bias, const int* __restrict__ batch,
                         float* __restrict__ h, float* __restrict__ bsum,
                         float* __restrict__ bcnt, int N) {
    int t = blockIdx.x * blockDim.x + threadIdx.x;
    if (t >= N * FF) return;
    int i = t / FF, f = t % FF;
    float v = fmaxf(outacc[t] / den[(size_t)i * HH + (f >> 2)] + bias[f], 0.f);
    h[t] = v;
    int b = batch[i];
    atomicAdd(&bsum[b * FF + f], v);
    if (f == 0) atomicAdd(&bcnt[b], 1.0f);
}

__global__ void k_bmean(const float* __restrict__ bsum, const float* __restrict__ bcnt,
                        float* __restrict__ mean) {
    int t = blockIdx.x * blockDim.x + threadIdx.x;
    if (t >= NB * FF) return;
    mean[t] = bsum[t] / bcnt[t / FF];
}

__global__ void k_center(const float* __restrict__ mean, const float* __restrict__ mscale,
                         const int* __restrict__ batch, float* __restrict__ h,
                         float* __restrict__ bvar, int N) {
    int t = blockIdx.x * blockDim.x + threadIdx.x;
    if (t >= N * FF) return;
    int i = t / FF, f = t % FF;
    int b = batch[i];
    float hc = h[t] - mean[b * FF + f] * mscale[f];
    h[t] = hc;
    atomicAdd(&bvar[b * FF + f], hc * hc);
}

__global__ void k_binvstd(const float* __restrict__ bvar, const float* __restrict__ bcnt,
                          float* __restrict__ invstd) {
    int t = blockIdx.x * blockDim.x + threadIdx.x;
    if (t >= NB * FF) return;
    invstd[t] = 1.0f / sqrtf(bvar[t] / bcnt[t / FF] + 1e-5f);
}

__global__ void k_norm(const float* __restrict__ invstd, const float* __restrict__ gnw,
                       const float* __restrict__ gnb, const int* __restrict__ batch,
                       float* __restrict__ h, int N) {
    int t = blockIdx.x * blockDim.x + threadIdx.x;
    if (t >= N * FF) return;
    int i = t / FF, f = t % FF;
    int b = batch[i];
    h[t] = gnw[f] * h[t] * invstd[b * FF + f] + gnb[f];
}

// ---- SAGE neighbor aggregation ---------------------------------------------
__global__ void k_sage_agg(const int* __restrict__ src, const int* __restrict__ dst,
                           const float* __restrict__ h, float* __restrict__ agg, int E) {
    int t = blockIdx.x * blockDim.x + threadIdx.x;
    if (t >= E * HH) return;
    int e = t / HH, q = t % HH;          // q: one float4 chunk of the 32 features
    int s = src[e], d = dst[e];
    const float4 hv = *(const float4*)(h + (size_t)s * FF + q * 4);
    float* ap = agg + (size_t)d * FF + q * 4;
    atomicAdd(&ap[0], hv.x);
    atomicAdd(&ap[1], hv.y);
    atomicAdd(&ap[2], hv.z);
    atomicAdd(&ap[3], hv.w);
}

__global__ void k_sageA(const float* __restrict__ agg, const float* __restrict__ h,
                        const float* __restrict__ deg, float* __restrict__ sageA, int N) {
    int t = blockIdx.x * blockDim.x + threadIdx.x;
    if (t >= N * FF) return;
    int i = t / FF, f = t % FF;
    sageA[(size_t)i * 64 + f]      = agg[t] / fmaxf(deg[i], 1.0f);
    sageA[(size_t)i * 64 + 32 + f] = h[t];
}

__global__ void k_wcat(const float* __restrict__ wl, const float* __restrict__ wr,
                       float* __restrict__ wcat) {
    int t = blockIdx.x * blockDim.x + threadIdx.x;
    if (t >= 64 * 64) return;
    int n = t / 64, k = t % 64;
    wcat[t] = (k < 32) ? wl[n * 32 + k] : wr[n * 32 + (k - 32)];
}

// ---- generic fp32 WMMA GEMM: C(M,Nout) = A(M,K) @ W(Nout,K)^T + bias -------
// one wave32 per 16x64 output tile: A fragment loaded once per K-step and fed
// to 4 independent V_WMMA_F32_16X16X4_F32 accumulator chains. Depth-1 software
// pipeline: next K-step's 5 fragments are fetched while the current 4 WMMAs
// run, so loads stay in flight instead of a per-WMMA s_wait_loadcnt 0.
// __launch_bounds__(...,1) lifts the occupancy-driven VGPR cap that previously
// forced single-register B staging. Requires M % 16 == 0, Nout % 64 == 0.
__global__ void __launch_bounds__(256, 1)
k_gemm_wmma(const float* __restrict__ A, int lda,
            const float* __restrict__ W,
            const float* __restrict__ bias,
            float* __restrict__ C, int ldc,
            int M, int K, int Nout, int doRelu) {
    int wavesPerBlk = blockDim.x >> 5;
    int wave = blockIdx.x * wavesPerBlk + (threadIdx.x >> 5);
    int tilesN = Nout >> 6;                    // 64-wide column tiles
    int total = (M >> 4) * tilesN;
    if (wave >= total) return;                 // uniform per wave: EXEC all-ones
    int tm = wave / tilesN, tn = wave % tilesN;
    int lane = threadIdx.x & 31;
    int half = lane >> 4;                      // K-pair select
    int r    = lane & 15;                      // A: row, B: col within 16-tile
    const float* Arow = A + (size_t)(tm * 16 + r) * lda;
    const float* W0 = W + (size_t)(tn * 64 + r) * K;
    const float* W1 = W0 + (size_t)16 * K;
    const float* W2 = W0 + (size_t)32 * K;
    const float* W3 = W0 + (size_t)48 * K;

    v8f acc[4] = {{}, {}, {}, {}};
    int ka = 2 * half;
    v2f a  = *(const v2f*)(Arow + ka);
    v2f b0 = *(const v2f*)(W0 + ka);
    v2f b1 = *(const v2f*)(W1 + ka);
    v2f b2 = *(const v2f*)(W2 + ka);
    v2f b3 = *(const v2f*)(W3 + ka);
    for (int k0 = 4; k0 < K; k0 += 4) {
        int kn = k0 + 2 * half;
        v2f an  = *(const v2f*)(Arow + kn);    // prefetch next step's fragments
        v2f b0n = *(const v2f*)(W0 + kn);
        v2f b1n = *(const v2f*)(W1 + kn);
        v2f b2n = *(const v2f*)(W2 + kn);
        v2f b3n = *(const v2f*)(W3 + kn);
        acc[0] = __builtin_amdgcn_wmma_f32_16x16x4_f32(false, a, false, b0, (short)0, acc[0], false, false);
        acc[1] = __builtin_amdgcn_wmma_f32_16x16x4_f32(false, a, false, b1, (short)0, acc[1], false, false);
        acc[2] = __builtin_amdgcn_wmma_f32_16x16x4_f32(false, a, false, b2, (short)0, acc[2], false, false);
        acc[3] = __builtin_amdgcn_wmma_f32_16x16x4_f32(false, a, false, b3, (short)0, acc[3], false, false);
        a = an; b0 = b0n; b1 = b1n; b2 = b2n; b3 = b3n;
    }
    acc[0] = __builtin_amdgcn_wmma_f32_16x16x4_f32(false, a, false, b0, (short)0, acc[0], false, false);
    acc[1] = __builtin_amdgcn_wmma_f32_16x16x4_f32(false, a, false, b1, (short)0, acc[1], false, false);
    acc[2] = __builtin_amdgcn_wmma_f32_16x16x4_f32(false, a, false, b2, (short)0, acc[2], false, false);
    acc[3] = __builtin_amdgcn_wmma_f32_16x16x4_f32(false, a, false, b3, (short)0, acc[3], false, false);

    int colBase = tn * 64 + r;
#pragma unroll
    for (int j = 0; j < 4; ++j) {
        float bv = bias[colBase + j * 16];
#pragma unroll
        for (int i = 0; i < 8; ++i) {
            int row = tm * 16 + i + 8 * half;
            float v = acc[j][i] + bv;
            if (doRelu) v = fmaxf(v, 0.f);
            C[(size_t)row * ldc + colBase + j * 16] = v;
        }
    }
}

// ---- pooling ---------------------------------------------------------------
__global__ void k_pool(const float* __restrict__ h2, const int* __restrict__ batch,
                       unsigned* __restrict__ g1ord, float* __restrict__ g2sum, int N) {
    int t = blockIdx.x * blockDim.x + threadIdx.x;
    if (t >= N * HIDN) return;
    int i = t / HIDN, f = t % HIDN;
    int b = batch[i];
    float v = h2[t];
    atomicMax(&g1ord[b * HIDN + f], ordEnc(v));
    atomicAdd(&g2sum[b * HIDN + f], v);
}

__global__ void k_gcat(const unsigned* __restrict__ g1ord, const float* __restrict__ g2sum,
                       const float* __restrict__ bcnt, float* __restrict__ g) {
    int t = blockIdx.x * blockDim.x + threadIdx.x;
    if (t >= NB * HIDN) return;
    int b = t / HIDN, f = t % HIDN;
    g[b * 128 + f]        = ordDec(g1ord[t]);
    g[b * 128 + 64 + f]   = g2sum[t] / bcnt[b];
}

// ---- fc3 (Nout=3, too thin for a 16-wide tile) -----------------------------
__global__ void k_fc3(const float* __restrict__ in, const float* __restrict__ w,
                      const float* __restrict__ b, float* __restrict__ out) {
    int t = blockIdx.x * blockDim.x + threadIdx.x;
    if (t >= NB * 3) return;
    int row = t / 3, j = t % 3;
    const float* ip = in + row * 512;
    const float* wp = w + j * 512;
    float acc = b[j];
    for (int k = 0; k < 512; ++k) acc += ip[k] * wp[k];
    out[t] = acc;
}

extern "C" void kernel_launch(void* const* d_in, const int* in_sizes, int n_in,
                              void* d_out, int out_size, void* d_ws, size_t ws_size,
                              hipStream_t stream) {
    const float* x        = (const float*)d_in[0];
    const int*   ei       = (const int*)d_in[1];
    const float* eattr    = (const float*)d_in[2];
    const int*   batch    = (const int*)d_in[3];
    const float* gat_wl   = (const float*)d_in[4];
    const float* gat_bl   = (const float*)d_in[5];
    const float* gat_wr   = (const float*)d_in[6];
    const float* gat_br   = (const float*)d_in[7];
    const float* gat_we   = (const float*)d_in[8];
    const float* gat_att  = (const float*)d_in[9];
    const float* gat_bias = (const float*)d_in[10];
    const float* gn_w     = (const float*)d_in[11];
    const float* gn_b     = (const float*)d_in[12];
    const float* gn_ms    = (const float*)d_in[13];
    const float* sage_wl  = (const float*)d_in[14];
    const float* sage_bl  = (const float*)d_in[15];
    const float* sage_wr  = (const float*)d_in[16];
    const float* fc1_w    = (const float*)d_in[17];
    const float* fc1_b    = (const float*)d_in[18];
    const float* fc2_w    = (const float*)d_in[19];
    const float* fc2_b    = (const float*)d_in[20];
    const float* fc3_w    = (const float*)d_in[21];
    const float* fc3_b    = (const float*)d_in[22];

    const int N = in_sizes[0];
    const int E = in_sizes[1] / 2;
    const int* src = ei;
    const int* dst = ei + E;

    // workspace carve-up (256B aligned slots)
    float* ws = (float*)d_ws;
    size_t off = 0;
    auto alloc = [&](size_t nfl) -> float* {
        float* p = ws + off;
        off += (nfl + 63) & ~(size_t)63;
        return p;
    };
    float*    xl      = alloc((size_t)N * FF);
    float*    xr      = alloc((size_t)N * FF);
    float*    deg     = alloc((size_t)N);
    float*    loopea  = alloc((size_t)N);
    unsigned* mord    = (unsigned*)alloc((size_t)N * HH);
    float*    den     = alloc((size_t)N * HH);
    float*    outacc  = alloc((size_t)N * FF);
    float*    hbuf    = alloc((size_t)N * FF);
    float*    agg     = alloc((size_t)N * FF);
    float*    sageA   = alloc((size_t)N * 64);
    float*    h2      = alloc((size_t)N * HIDN);
    float*    wcat    = alloc(64 * 64);
    float*    bsum    = alloc(NB * FF);
    float*    bcnt    = alloc(NB);
    float*    bmean   = alloc(NB * FF);
    float*    bvar    = alloc(NB * FF);
    float*    binvstd = alloc(NB * FF);
    unsigned* g1ord   = (unsigned*)alloc(NB * HIDN);
    float*    g2sum   = alloc(NB * HIDN);
    float*    g       = alloc(NB * 128);
    float*    buf1    = alloc(NB * 1024);
    float*    buf2    = alloc(NB * 512);

    // zero all accumulators (ordered-uint 0 == -inf for the max reductions)
    hipMemsetAsync(d_ws, 0, off * sizeof(float), stream);

    const int T = 256;
    auto nb = [](long long n, int t) { return (unsigned)((n + t - 1) / t); };

    k_deg<<<nb(E, T), T, 0, stream>>>(dst, eattr, deg, loopea, E);
    k_node_prep<<<nb((long long)N * FF, T), T, 0, stream>>>(
        x, gat_wl, gat_bl, gat_wr, gat_br, deg, xl, xr, loopea, N);
    k_alpha_max<<<nb((long long)(E + N) * HH, T), T, 0, stream>>>(
        src, dst, eattr, loopea, xl, xr, gat_we, gat_att, mord, E, N);
    k_soft_acc<<<nb((long long)(E + N) * HH, T), T, 0, stream>>>(
        src, dst, eattr, loopea, xl, xr, gat_we, gat_att, mord, den, outacc, E, N);
    k_h_relu<<<nb((long long)N * FF, T), T, 0, stream>>>(
        outacc, den, gat_bias, batch, hbuf, bsum, bcnt, N);
    k_bmean<<<nb(NB * FF, T), T, 0, stream>>>(bsum, bcnt, bmean);
    k_center<<<nb((long long)N * FF, T), T, 0, stream>>>(bmean, gn_ms, batch, hbuf, bvar, N);
    k_binvstd<<<nb(NB * FF, T), T, 0, stream>>>(bvar, bcnt, binvstd);
    k_norm<<<nb((long long)N * FF, T), T, 0, stream>>>(binvstd, gn_w, gn_b, batch, hbuf, N);
    k_sage_agg<<<nb((long long)E * HH, T), T, 0, stream>>>(src, dst, hbuf, agg, E);
    k_sageA<<<nb((long long)N * FF, T), T, 0, stream>>>(agg, hbuf, deg, sageA, N);
    k_wcat<<<nb(64 * 64, T), T, 0, stream>>>(sage_wl, sage_wr, wcat);

    // SAGE GEMM: (N,64) @ (64,64)^T, relu   [wave32 WMMA f32 16x16x4, 16x64 tile]
    {
        long long waves = (long long)(N / 16) * (64 / 64);
        k_gemm_wmma<<<nb(waves * 32, T), T, 0, stream>>>(sageA, 64, wcat, sage_bl,
                                                         h2, 64, N, 64, 64, 1);
    }
    k_pool<<<nb((long long)N * HIDN, T), T, 0, stream>>>(h2, batch, g1ord, g2sum, N);
    k_gcat<<<nb(NB * HIDN, T), T, 0, stream>>>(g1ord, g2sum, bcnt, g);

    // fc1: (64,128)@(128,1024)^T relu ; fc2: (64,1024)@(1024,512)^T relu
    {
        long long waves1 = (long long)(NB / 16) * (1024 / 64);
        k_gemm_wmma<<<nb(waves1 * 32, T), T, 0, stream>>>(g, 128, fc1_w, fc1_b,
                                                          buf1, 1024, NB, 128, 1024, 1);
        long long waves2 = (long long)(NB / 16) * (512 / 64);
        k_gemm_wmma<<<nb(waves2 * 32, T), T, 0, stream>>>(buf1, 1024, fc2_w, fc2_b,
                                                          buf2, 512, NB, 1024, 512, 1);
    }
    k_fc3<<<1, 256, 0, stream>>>(buf2, fc3_w, fc3_b, (float*)d_out);
}